// Conv_Embedding_1_39462159515868
// MI455X (gfx1250) — compile-verified
//
#include <hip/hip_runtime.h>

typedef __attribute__((ext_vector_type(2))) float v2f;
typedef __attribute__((ext_vector_type(8))) float v8f;

#define D    128
#define TILE 16

// ---------------------------------------------------------------------------
// H0[n,128] = E[n,128] @ W[128,128] + bias   (fp32 WMMA 16x16x4)
// One block = 16 output rows; 8 waves, wave w owns columns [16w, 16w+16).
// A tile staged through LDS; K looped 0..124 step 4.
// ---------------------------------------------------------------------------
__global__ __launch_bounds__(256) void gemm_wmma_kernel(
    const float* __restrict__ E, const float* __restrict__ W,
    const float* __restrict__ bias, float* __restrict__ H0)
{
    __shared__ float As[TILE * D];                       // 8 KB
    const int row0 = blockIdx.x * TILE;
    const int tid  = threadIdx.x;

    // Cooperative stage of the 16x128 A tile (2048 floats = 512 float4)
    const float4* esrc = (const float4*)(E + (size_t)row0 * D);
    float4* adst = (float4*)As;
    adst[tid]       = esrc[tid];
    adst[tid + 256] = esrc[tid + 256];
    __syncthreads();

    const int wave = tid >> 5;
    const int lane = tid & 31;
    const int n0   = wave * TILE;
    const int m    = lane & 15;    // A-row within tile / B-col within tile
    const int half = lane >> 4;    // selects K-pair within the K=4 step

    v8f acc = {};
#pragma unroll
    for (int kk = 0; kk < D / 4; ++kk) {
        const int k0 = kk * 4 + 2 * half;
        // A fragment: lane holds A[m][k0], A[m][k0+1]  (ISA 16x4 f32 layout)
        v2f a = *(const v2f*)(&As[m * D + k0]);
        // B fragment: lane holds B[k0][n0+m], B[k0+1][n0+m]
        v2f b;
        b.x = W[(size_t)k0 * D + n0 + m];
        b.y = W[(size_t)(k0 + 1) * D + n0 + m];
        acc = __builtin_amdgcn_wmma_f32_16x16x4_f32(
            /*neg_a=*/false, a, /*neg_b=*/false, b,
            /*c_mod=*/(short)0, acc, /*reuse_a=*/false, /*reuse_b=*/false);
    }

    // C/D layout: VGPR j -> M = j + 8*half, N = lane&15
    const int   n     = n0 + m;
    const float bs    = bias[n];
    const int   rbase = row0 + half * 8;
#pragma unroll
    for (int j = 0; j < 8; ++j)
        H0[(size_t)(rbase + j) * D + n] = acc[j] + bs;
}

// ---------------------------------------------------------------------------
__global__ void zero_kernel(float4* __restrict__ p, long n4)
{
    long i = (long)blockIdx.x * blockDim.x + threadIdx.x;
    if (i < n4) p[i] = make_float4(0.f, 0.f, 0.f, 0.f);
}

// ---------------------------------------------------------------------------
// Wave per edge: H1[rows[e], :] += vals[e] * H0[cols[e], :]
// ---------------------------------------------------------------------------
__global__ __launch_bounds__(256) void spmm_kernel(
    const float* __restrict__ H0, const float* __restrict__ vals,
    const int* __restrict__ rows, const int* __restrict__ cols,
    float* __restrict__ H1, int nedges)
{
    int e = (int)(((long)blockIdx.x * blockDim.x + threadIdx.x) >> 5);
    if (e >= nedges) return;
    int   lane = threadIdx.x & 31;
    int   r = rows[e], c = cols[e];
    float v = vals[e];
    float4 m = ((const float4*)(H0 + (size_t)c * D))[lane];
    float* dst = H1 + (size_t)r * D + lane * 4;
    atomicAdd(dst + 0, m.x * v);
    atomicAdd(dst + 1, m.y * v);
    atomicAdd(dst + 2, m.z * v);
    atomicAdd(dst + 3, m.w * v);
}

// ---------------------------------------------------------------------------
// Wave per node: relu then LayerNorm over 128 features (wave32 reductions).
// ---------------------------------------------------------------------------
__global__ __launch_bounds__(256) void relu_ln_kernel(
    float* __restrict__ H, const float* __restrict__ gamma,
    const float* __restrict__ beta, int n)
{
    int node = (int)(((long)blockIdx.x * blockDim.x + threadIdx.x) >> 5);
    if (node >= n) return;
    int lane = threadIdx.x & 31;

    float4* rowp = (float4*)(H + (size_t)node * D);
    float4  h    = rowp[lane];
    h.x = fmaxf(h.x, 0.f); h.y = fmaxf(h.y, 0.f);
    h.z = fmaxf(h.z, 0.f); h.w = fmaxf(h.w, 0.f);

    float s = h.x + h.y + h.z + h.w;
#pragma unroll
    for (int off = 16; off > 0; off >>= 1) s += __shfl_xor(s, off, 32);
    float mu = s * (1.0f / D);

    float dx = h.x - mu, dy = h.y - mu, dz = h.z - mu, dw = h.w - mu;
    float var = dx * dx + dy * dy + dz * dz + dw * dw;
#pragma unroll
    for (int off = 16; off > 0; off >>= 1) var += __shfl_xor(var, off, 32);
    float rs = rsqrtf(var * (1.0f / D) + 1e-5f);

    float4 g = ((const float4*)gamma)[lane];
    float4 b = ((const float4*)beta)[lane];
    float4 o;
    o.x = dx * rs * g.x + b.x;
    o.y = dy * rs * g.y + b.y;
    o.z = dz * rs * g.z + b.z;
    o.w = dw * rs * g.w + b.w;
    rowp[lane] = o;
}

// ---------------------------------------------------------------------------
// Wave per batch row: masked gather  out[b,:] = (1<=x[b]<=n) ? H[x[b]-1,:] : 0
// ---------------------------------------------------------------------------
__global__ __launch_bounds__(256) void gather_kernel(
    const int* __restrict__ x, const float* __restrict__ H,
    float* __restrict__ out, int batch, int n)
{
    int b = (int)(((long)blockIdx.x * blockDim.x + threadIdx.x) >> 5);
    if (b >= batch) return;
    int lane = threadIdx.x & 31;
    int xi = x[b];
    float4 o = make_float4(0.f, 0.f, 0.f, 0.f);
    if (xi >= 1 && xi <= n)
        o = ((const float4*)(H + (size_t)(xi - 1) * D))[lane];
    ((float4*)(out + (size_t)b * D))[lane] = o;
}

// ---------------------------------------------------------------------------
extern "C" void kernel_launch(void* const* d_in, const int* in_sizes, int n_in,
                              void* d_out, int out_size, void* d_ws, size_t ws_size,
                              hipStream_t stream)
{
    const int*   x     = (const int*)d_in[0];
    const float* emb   = (const float*)d_in[1];
    const float* W     = (const float*)d_in[2];
    const float* bias  = (const float*)d_in[3];
    const float* vals  = (const float*)d_in[4];
    const int*   rows  = (const int*)d_in[5];
    const int*   cols  = (const int*)d_in[6];
    const float* gamma = (const float*)d_in[7];
    const float* beta  = (const float*)d_in[8];
    float*       out   = (float*)d_out;

    const int batch = in_sizes[0];
    const int n     = in_sizes[1] / D;   // 100000, divisible by 16
    const int ned   = in_sizes[4];

    float* H0 = (float*)d_ws;            // [n,128] dense GEMM result
    float* H1 = H0 + (size_t)n * D;      // [n,128] SpMM accumulator

    // Zero the accumulator (ws is not re-initialized between replays).
    long n4 = (long)n * D / 4;
    zero_kernel<<<(unsigned)((n4 + 255) / 256), 256, 0, stream>>>((float4*)H1, n4);

    // Dense GEMM on the WMMA pipe.
    gemm_wmma_kernel<<<n / TILE, 256, 0, stream>>>(emb, W, bias, H0);

    // Sparse scatter-add (wave per edge).
    spmm_kernel<<<(ned + 7) / 8, 256, 0, stream>>>(H0, vals, rows, cols, H1, ned);

    // relu + LayerNorm (wave per node).
    relu_ln_kernel<<<(n + 7) / 8, 256, 0, stream>>>(H1, gamma, beta, n);

    // Masked gather into the output table (wave per batch row).
    gather_kernel<<<(batch + 7) / 8, 256, 0, stream>>>(x, H1, out, batch, n);
}